// Net_55405078118494
// MI455X (gfx1250) — compile-verified
//
#include <hip/hip_runtime.h>

#define N_NODES  50000
#define N_EDGES  400000
#define NODE_DIM 32

typedef __attribute__((ext_vector_type(2))) float v2f;
typedef __attribute__((ext_vector_type(8))) float v8f;

// Zero output accumulator and per-node edge counts (must happen every launch:
// harness poisons buffers once and replays the captured graph).
__global__ void gnn_init_kernel(float* __restrict__ out, float* __restrict__ cnt) {
    int i = blockIdx.x * blockDim.x + threadIdx.x;
    if (i < N_NODES * NODE_DIM) out[i] = 0.0f;
    if (i < N_NODES) cnt[i] = 0.0f;
}

// One wave32 per edge: y = x_src (1x32) * A_e (32x32), scatter-add into out[dst].
// WMMA mapping (V_WMMA_F32_16X16X4_F32, D = A*B + C):
//   A (16x4): x[k0..k0+3] broadcast into all 16 M rows.
//     Layout: lanes 0-15 are M=0..15; VGPR0 holds K0 (lanes 0-15) / K2 (16-31),
//             VGPR1 holds K1 / K3  -> a.x = x[k0+2*half], a.y = x[k0+2*half+1].
//   B (4x16): edge matrix chunk, element (K=d, N=k) = A_e[d][k].
//     Layout mirror: lane n holds column N=n; half selects K parity pair.
//   C (16x16): every M row identical == desired 1x16 result slice; read row M=0
//     (VGPR0 of the accumulator, lanes 0-15).
__global__ __launch_bounds__(256) void gnn_edge_kernel(
    const float* __restrict__ node_states,
    const float* __restrict__ a_in,
    const long long* __restrict__ edge_index,
    float* __restrict__ out,
    float* __restrict__ cnt) {

    const int wave = (blockIdx.x * blockDim.x + threadIdx.x) >> 5;
    if (wave >= N_EDGES) return;                 // wave-uniform; EXEC stays all-1s
    const int lane = threadIdx.x & 31;
    const int half = lane >> 4;                  // 0: lanes 0-15, 1: lanes 16-31
    const int n    = lane & 15;                  // N column within half

    const int src = (int)edge_index[2 * (size_t)wave + 0];
    const int dst = (int)edge_index[2 * (size_t)wave + 1];

    // Gather source node vector: lane i holds x[i]. Node table (6.4 MB) is
    // L2-resident; default RT temporal hint keeps it cached.
    const float xv = node_states[(size_t)src * NODE_DIM + lane];

    const float* __restrict__ Ae = a_in + (size_t)wave * (NODE_DIM * NODE_DIM);

    v8f c_lo = {};   // output columns 0..15
    v8f c_hi = {};   // output columns 16..31

#pragma unroll
    for (int k0 = 0; k0 < NODE_DIM; k0 += 4) {
        const int r0 = k0 + 2 * half;            // K row pair handled by this half

        // A operand: broadcast x chunk per the 16x4 f32 A layout.
        v2f a;
        a.x = __shfl(xv, r0,     32);
        a.y = __shfl(xv, r0 + 1, 32);

        // B operand: stream the edge matrix with NT hint (1.64 GB, zero reuse;
        // keep it out of the 192 MB L2 so node/out stay resident).
        v2f b0, b1;
        b0.x = __builtin_nontemporal_load(Ae + (size_t)r0       * NODE_DIM + n);
        b0.y = __builtin_nontemporal_load(Ae + (size_t)(r0 + 1) * NODE_DIM + n);
        b1.x = __builtin_nontemporal_load(Ae + (size_t)r0       * NODE_DIM + 16 + n);
        b1.y = __builtin_nontemporal_load(Ae + (size_t)(r0 + 1) * NODE_DIM + 16 + n);

        c_lo = __builtin_amdgcn_wmma_f32_16x16x4_f32(
            /*neg_a=*/false, a, /*neg_b=*/false, b0,
            /*c_mod=*/(short)0, c_lo, /*reuse_a=*/false, /*reuse_b=*/false);
        c_hi = __builtin_amdgcn_wmma_f32_16x16x4_f32(
            /*neg_a=*/false, a, /*neg_b=*/false, b1,
            /*c_mod=*/(short)0, c_hi, /*reuse_a=*/false, /*reuse_b=*/false);
    }

    // Row M=0 of C/D lives in accumulator VGPR0 on lanes 0-15.
    float* outrow = out + (size_t)dst * NODE_DIM;
    if (lane < 16) {
        atomicAdd(outrow + lane,      c_lo[0]);   // global_atomic_add_f32 (no-return)
        atomicAdd(outrow + 16 + lane, c_hi[0]);
    }
    if (lane == 0) atomicAdd(cnt + dst, 1.0f);
}

__global__ void gnn_finalize_kernel(float* __restrict__ out,
                                    const float* __restrict__ cnt) {
    int i = blockIdx.x * blockDim.x + threadIdx.x;
    if (i < N_NODES * NODE_DIM) {
        out[i] = out[i] / fmaxf(cnt[i >> 5], 1.0f);
    }
}

extern "C" void kernel_launch(void* const* d_in, const int* in_sizes, int n_in,
                              void* d_out, int out_size, void* d_ws, size_t ws_size,
                              hipStream_t stream) {
    const float*     node_states = (const float*)d_in[0];      // (50000, 32) f32
    const float*     a_in        = (const float*)d_in[1];      // (400000, 32, 32) f32
    const long long* edge_index  = (const long long*)d_in[2];  // (400000, 2) int64
    float* out = (float*)d_out;                                // (50000, 32) f32
    float* cnt = (float*)d_ws;                                 // 50000 f32 scratch

    const int total_out = N_NODES * NODE_DIM;

    gnn_init_kernel<<<(total_out + 255) / 256, 256, 0, stream>>>(out, cnt);

    // 400000 edges * 32 lanes/edge / 256 threads = 50000 blocks exactly.
    gnn_edge_kernel<<<(N_EDGES * 32) / 256, 256, 0, stream>>>(
        node_states, a_in, edge_index, out, cnt);

    gnn_finalize_kernel<<<(total_out + 255) / 256, 256, 0, stream>>>(out, cnt);
}